// ControlSystem_88716844466958
// MI455X (gfx1250) — compile-verified
//
#include <hip/hip_runtime.h>
#include <math.h>

// ---- Model constants (match the Python reference, folded to f32) ----
// MC=0.2  MP=0.5  L=0.5  G=9.80364  TF=20  NSTEPS=200  DT=0.1
#define NSTEPS 200
#define K_DT        0.1f
#define K_MC        0.2f
#define K_MP        0.5f
#define K_G         9.80364f
#define K_MPGL      (0.5f * 9.80364f * 0.5f)    // MP*G*L       = 2.45091
#define K_HALFMTOT  (0.5f * (0.2f + 0.5f))      // 0.5*(MC+MP)  = 0.35
#define K_MPL       (0.5f * 0.5f)               // MP*L         = 0.25
#define K_HALFMPLL  (0.5f * 0.5f * 0.5f * 0.5f) // 0.5*MP*L*L   = 0.0625
#define K_MTOTG     ((0.2f + 0.5f) * 9.80364f)  // (MC+MP)*G    = 6.862548

// Branch-free Cody-Waite sincos, accurate to ~1-2 ulp for |x| <~ 1e5 rad.
// (Angle magnitude is physically bounded far below that over the 20 s
//  rollout; inf/NaN states propagate to NaN exactly like libm.)
__device__ __forceinline__ void fast_sincosf(float x, float* sout, float* cout) {
    // q = round(x / (pi/2))
    const float q = rintf(x * 0.63661975f);           // v_rndne_f32
    // 3-term FMA Cody-Waite reduction: r = x - q*(pi/2), exact to ~1 ulp
    float r = fmaf(q, -1.5707962513e+00f, x);
    r = fmaf(q, -7.5497894159e-08f, r);
    r = fmaf(q, -5.3903029470e-15f, r);
    const int n = (int)q;                             // v_cvt_i32_f32

    const float r2 = r * r;
    // sin(r), |r| <= pi/4  (Cephes minimax)
    float sp = fmaf(r2, -1.9515296e-04f, 8.3321609e-03f);
    sp = fmaf(r2, sp, -1.6666655e-01f);
    const float s = fmaf(r2 * r, sp, r);
    // cos(r), |r| <= pi/4
    float cp = fmaf(r2, 2.4433157e-05f, -1.3887316e-03f);
    cp = fmaf(r2, cp, 4.1666646e-02f);
    const float c = fmaf(r2 * r2, cp, fmaf(r2, -0.5f, 1.0f));

    // Quadrant: odd q swaps sin/cos; signs via sign-bit XOR (branchless).
    const bool swap = (n & 1) != 0;
    const float ts = swap ? c : s;                    // v_cndmask
    const float tc = swap ? s : c;
    const int ssign = (n & 2) << 30;                  // 0x80000000 if n&2
    const int csign = ((n + 1) & 2) << 30;
    *sout = __int_as_float(__float_as_int(ts) ^ ssign);
    *cout = __int_as_float(__float_as_int(tc) ^ csign);
}

__device__ __forceinline__ float stage_loss(float x, float xd, float thd,
                                            float cos_t, float u) {
    return 0.05f * x * x
         + K_MPGL * (1.0f - cos_t)
         + K_HALFMTOT * xd * xd
         + K_MPL * xd * thd * cos_t
         + K_HALFMPLL * thd * thd
         + 0.1f * u * u;
}

__global__ __launch_bounds__(256) void cartpole_rollout_kernel(
    const float* __restrict__ x_in,   // (B, 4) row-major
    const float* __restrict__ W_in,   // (1, 4)
    float* __restrict__ out,          // (B, 1)
    int B)
{
    const int i = blockIdx.x * blockDim.x + threadIdx.x;
    if (i >= B) return;

    // Per-thread state: one coalesced 16B load (global_load_b128).
    const float4 s0 = *reinterpret_cast<const float4*>(x_in + 4 * i);
    float sx  = s0.x;   // cart position
    float th  = s0.y;   // pole angle
    float xd  = s0.z;   // cart velocity
    float thd = s0.w;   // pole angular velocity

    // Uniform weights -> scalar loads (s_load_b128), broadcast to all lanes.
    const float w0 = W_in[0];
    const float w1 = W_in[1];
    const float w2 = W_in[2];
    const float w3 = W_in[3];

    float acc = 0.0f;

    #pragma unroll 4
    for (int t = 0; t < NSTEPS; ++t) {
        // Controller: u = s . w  (uniform-weight dot -> 4 FMAs)
        const float u = sx * w0 + th * w1 + xd * w2 + thd * w3;

        float sin_t, cos_t;
        fast_sincosf(th, &sin_t, &cos_t);

        // Stage loss at the CURRENT state (before the Euler step).
        acc += stage_loss(sx, xd, thd, cos_t, u);

        // Dynamics
        const float thd2  = thd * thd;
        const float denom = fmaf(K_MP * sin_t, sin_t, K_MC);
        const float inv   = 1.0f / denom;               // one IEEE divide/step
        const float xdd   = fmaf(K_MP * sin_t,
                                 fmaf(0.5f, thd2, -K_G * cos_t), u) * inv;
        // 1/(L*denom) == 2*inv up to 1 ulp (L = 0.5, power of two)
        const float thdd  = (fmaf(-K_MPL * thd2, sin_t * cos_t,
                                  K_MTOTG * sin_t) - u * cos_t) * (2.0f * inv);

        // Euler update: position/angle advance with OLD velocities.
        sx  = fmaf(K_DT, xd,   sx);
        th  = fmaf(K_DT, thd,  th);
        xd  = fmaf(K_DT, xdd,  xd);
        thd = fmaf(K_DT, thdd, thd);
    }

    // Terminal stage loss at s_final.
    {
        const float u = sx * w0 + th * w1 + xd * w2 + thd * w3;
        float sin_t, cos_t;
        fast_sincosf(th, &sin_t, &cos_t);   // sin half is dead-code-eliminated
        acc += stage_loss(sx, xd, thd, cos_t, u);
    }

    out[i] = acc / 201.0f;  // acc / (NSTEPS + 1)
}

extern "C" void kernel_launch(void* const* d_in, const int* in_sizes, int n_in,
                              void* d_out, int out_size, void* d_ws, size_t ws_size,
                              hipStream_t stream) {
    const float* x = (const float*)d_in[0];  // (B,4) fp32
    const float* W = (const float*)d_in[1];  // (1,4) fp32
    float* out     = (float*)d_out;          // (B,1) fp32

    const int B = in_sizes[0] / 4;
    const int block = 256;                   // 8 wave32s per block
    const int grid  = (B + block - 1) / block;
    cartpole_rollout_kernel<<<grid, block, 0, stream>>>(x, W, out, B);
}